// ContinuousActorCritic_70858370449713
// MI455X (gfx1250) — compile-verified
//
#include <hip/hip_runtime.h>
#include <hip/hip_bf16.h>
#include <cstddef>

// ---------------------------------------------------------------------------
// ContinuousActorCritic forward on MI455X (gfx1250, wave32, WMMA)
//
// Roofline: x read = 256MB (~11us @23.3TB/s). FLOPs = 34G (xproj, parallel)
// + 69G (sequential scan, 2048 dependent steps). Scan is the critical path
// and is LDS-bandwidth bound for f32 WMMA (4 FLOP/byte), so: W_hh resident
// in 320KB WGP LDS, half its fragments hoisted to VGPRs (loop-invariant),
// each A fragment reused across 2 N-tiles, xp C-init software-pipelined.
// ---------------------------------------------------------------------------

typedef float v2f __attribute__((ext_vector_type(2)));
typedef float v8f __attribute__((ext_vector_type(8)));

#define ALPHA 0.2f

constexpr int Bt = 256;   // batch
constexpr int Tt = 2048;  // timesteps
constexpr int It = 128;   // input dim
constexpr int Ht = 256;   // hidden dim
constexpr int At = 8;     // actions

// LDS row pads: stride % 64 banks == 4 -> conflict-free strided b64 fragment
// loads; stride*4 % 16 == 0 keeps float4 staging stores aligned.
constexpr int WP1 = 132;  // W_ih row stride (128+4)
constexpr int XP1 = 132;  // x-tile row stride
constexpr int WP2 = 260;  // W_hh row stride (256+4)
constexpr int HP2 = 260;  // h-tile row stride

// ---------------------------------------------------------------------------
// Phase 1: xp[t,b,:] = x[b,t,:] @ W_ih^T  for a chunk of Tc timesteps.
// Block = 64 rows (4 M-tiles) x all 256 cols; 16 waves, wave = one N-tile,
// 4 accumulators -> each B fragment (W_ih) reused 4x from LDS.
// ---------------------------------------------------------------------------
__global__ __launch_bounds__(512)
void xproj_kernel(const float* __restrict__ x, const float* __restrict__ W_ih,
                  float* __restrict__ xp, int t0) {
  extern __shared__ float lds[];
  float* w_lds = lds;                 // [256][WP1]
  float* x_lds = lds + 256 * WP1;     // [64][XP1]

  const int tid  = threadIdx.x;
  const int lane = tid & 31;
  const int wave = tid >> 5;          // 0..15 == N-tile
  const int r0   = blockIdx.x * 64;   // first (t_local*B + b) row of this block

  // Stage W_ih [256][128] -> LDS (L2-resident, 135KB)
  for (int i = tid; i < 256 * 32; i += 512) {
    const int r = i >> 5, c = (i & 31) << 2;
    *(float4*)(w_lds + r * WP1 + c) = *(const float4*)(W_ih + r * It + c);
  }
  // Stage x tile: rows r0..r0+63; row r -> t = t0 + r/256, b = r%256
  for (int i = tid; i < 64 * 32; i += 512) {
    const int rl = i >> 5, c = (i & 31) << 2;
    const int r = r0 + rl;
    const int t = t0 + (r >> 8);
    const int b = r & 255;
    *(float4*)(x_lds + rl * XP1 + c) =
        *(const float4*)(x + ((size_t)b * Tt + t) * It + c);
  }
  __syncthreads();

  const int n0      = wave << 4;
  const int halfsel = (lane >> 4) << 1;   // K sub-select: 0 | 2
  const int mrow    = lane & 15;
  const float* brow = w_lds + (n0 + mrow) * WP1 + halfsel; // B[k][n]=W_ih[n][k]
  const float* arow = x_lds + mrow * XP1 + halfsel;

  v8f acc[4] = {v8f{}, v8f{}, v8f{}, v8f{}};
#pragma unroll 4
  for (int k = 0; k < It; k += 4) {
    const v2f bf = *(const v2f*)(brow + k);
#pragma unroll
    for (int ms = 0; ms < 4; ++ms) {
      const v2f af = *(const v2f*)(arow + ms * 16 * XP1 + k);
      acc[ms] = __builtin_amdgcn_wmma_f32_16x16x4_f32(
          false, af, false, bf, (short)0, acc[ms], false, false);
    }
  }

  // C layout: VGPR r -> M = r + 8*(lane>=16), N = lane&15
  const int Moff = (lane >> 4) << 3;
  const int ncol = n0 + (lane & 15);
#pragma unroll
  for (int ms = 0; ms < 4; ++ms) {
#pragma unroll
    for (int r = 0; r < 8; ++r) {
      const int row = r0 + ms * 16 + Moff + r;
      xp[(size_t)row * Ht + ncol] = acc[ms][r];   // coalesced 64B / lane-half
    }
  }
}

// ---------------------------------------------------------------------------
// Phase 2: sequential scan over Tc steps.
// 16 blocks (16-row B-slices), 8 waves x 2 N-tiles. W_hh in LDS (260KB);
// K in [0,128) W_hh fragments hoisted to 128 VGPRs (invariant over t).
// pre = (xp tile, software-pipelined C-init) + h @ W_hh^T
// h   = 0.8h + 0.2 relu(pre); lane-owned h carried in registers.
// ---------------------------------------------------------------------------
__global__ __launch_bounds__(256)
void scan_kernel(const float* __restrict__ xp, const float* __restrict__ W_hh,
                 float* __restrict__ hstate, int Tc) {
  extern __shared__ float lds[];
  float* w_lds = lds;                 // [256][WP2]
  float* h_lds = lds + 256 * WP2;     // [16][HP2]

  const int tid  = threadIdx.x;
  const int lane = tid & 31;
  const int wave = tid >> 5;          // 0..7
  const int mb   = blockIdx.x;        // B-slice

  // Stage W_hh [256][256] -> LDS once; amortized over Tc steps
  for (int i = tid; i < 256 * 64; i += 256) {
    const int r = i >> 6, c = (i & 63) << 2;
    *(float4*)(w_lds + r * WP2 + c) = *(const float4*)(W_hh + r * Ht + c);
  }
  // Stage this block's h rows
  for (int i = tid; i < 16 * 64; i += 256) {
    const int rl = i >> 6, c = (i & 63) << 2;
    *(float4*)(h_lds + rl * HP2 + c) =
        *(const float4*)(hstate + (size_t)(mb * 16 + rl) * Ht + c);
  }
  __syncthreads();

  const int halfsel = (lane >> 4) << 1;   // fragment K sub-select: 0 | 2
  const int mrow    = lane & 15;
  const int Moff    = (lane >> 4) << 3;   // C rows: 0..7 | 8..15
  const int nc      = lane & 15;
  const int n0a     = wave << 4;          // N-tile a: 0..7
  const int n0b     = (wave + 8) << 4;    // N-tile b: 8..15
  const float* arow  = h_lds + mrow * HP2 + halfsel;
  const float* browa = w_lds + (n0a + mrow) * WP2 + halfsel;
  const float* browb = w_lds + (n0b + mrow) * WP2 + halfsel;

  // Hoist K in [0,128) B fragments into registers (invariant across steps)
  v2f brega[32], bregb[32];
#pragma unroll
  for (int i = 0; i < 32; ++i) {
    brega[i] = *(const v2f*)(browa + 4 * i);
    bregb[i] = *(const v2f*)(browb + 4 * i);
  }

  // Lane-owned h entries carried in registers across steps
  float hca[8], hcb[8];
#pragma unroll
  for (int r = 0; r < 8; ++r) {
    hca[r] = h_lds[(Moff + r) * HP2 + n0a + nc];
    hcb[r] = h_lds[(Moff + r) * HP2 + n0b + nc];
  }

  // xp addressing: element (t, r) at xpa[(t*Bt + r)*Ht]
  const float* xpa = xp + (size_t)(mb * 16 + Moff) * Ht + n0a + nc;
  const float* xpb = xp + (size_t)(mb * 16 + Moff) * Ht + n0b + nc;

  // Prefetch C-init for t = 0
  float nxa[8], nxb[8];
#pragma unroll
  for (int r = 0; r < 8; ++r) {
    nxa[r] = xpa[(size_t)r * Ht];
    nxb[r] = xpb[(size_t)r * Ht];
  }

  for (int t = 0; t < Tc; ++t) {
    v8f acc0, acc1;
#pragma unroll
    for (int r = 0; r < 8; ++r) { acc0[r] = nxa[r]; acc1[r] = nxb[r]; }

    // Prefetch next step's xp tile; latency hidden under this step's K-loop
    if (t + 1 < Tc) {
      const float* pa = xpa + (size_t)(t + 1) * Bt * Ht;
      const float* pb = xpb + (size_t)(t + 1) * Bt * Ht;
#pragma unroll
      for (int r = 0; r < 8; ++r) {
        nxa[r] = pa[(size_t)r * Ht];
        nxb[r] = pb[(size_t)r * Ht];
      }
    }

    // K in [0,128): B from registers, A from LDS (shared across both tiles)
#pragma unroll
    for (int i = 0; i < 32; ++i) {
      const v2f af = *(const v2f*)(arow + 4 * i);
      acc0 = __builtin_amdgcn_wmma_f32_16x16x4_f32(
          false, af, false, brega[i], (short)0, acc0, false, false);
      acc1 = __builtin_amdgcn_wmma_f32_16x16x4_f32(
          false, af, false, bregb[i], (short)0, acc1, false, false);
    }
    // K in [128,256): B from LDS
#pragma unroll 8
    for (int k = 128; k < 256; k += 4) {
      const v2f af = *(const v2f*)(arow + k);
      const v2f b0 = *(const v2f*)(browa + k);
      const v2f b1 = *(const v2f*)(browb + k);
      acc0 = __builtin_amdgcn_wmma_f32_16x16x4_f32(
          false, af, false, b0, (short)0, acc0, false, false);
      acc1 = __builtin_amdgcn_wmma_f32_16x16x4_f32(
          false, af, false, b1, (short)0, acc1, false, false);
    }

    // Leaky-ReLU blend on register-carried h
#pragma unroll
    for (int r = 0; r < 8; ++r) {
      const float p0 = acc0[r], p1 = acc1[r];
      hca[r] = (1.0f - ALPHA) * hca[r] + ALPHA * (p0 > 0.0f ? p0 : 0.0f);
      hcb[r] = (1.0f - ALPHA) * hcb[r] + ALPHA * (p1 > 0.0f ? p1 : 0.0f);
    }
    __syncthreads();   // everyone done reading all of h for this step
#pragma unroll
    for (int r = 0; r < 8; ++r) {
      h_lds[(Moff + r) * HP2 + n0a + nc] = hca[r];
      h_lds[(Moff + r) * HP2 + n0b + nc] = hcb[r];
    }
    __syncthreads();   // h fully updated before next step's reads
  }

  // Persist h slice for next chunk / heads
  for (int i = tid; i < 16 * 64; i += 256) {
    const int rl = i >> 6, c = (i & 63) << 2;
    *(float4*)(hstate + (size_t)(mb * 16 + rl) * Ht + c) =
        *(const float4*)(h_lds + rl * HP2 + c);
  }
}

// ---------------------------------------------------------------------------
// Heads: actor = h @ W_actor^T, critic = h @ W_critic^T, new_hx = h (tiny)
// out layout: actor[256*8] | critic[256] | hx[256*256]
// ---------------------------------------------------------------------------
__global__ __launch_bounds__(256)
void heads_kernel(const float* __restrict__ hstate,
                  const float* __restrict__ W_actor,
                  const float* __restrict__ W_critic,
                  float* __restrict__ out) {
  __shared__ float hrow[Ht];
  const int b = blockIdx.x;
  const int k = threadIdx.x;
  const float hv = hstate[(size_t)b * Ht + k];
  out[Bt * At + Bt + (size_t)b * Ht + k] = hv;   // new_hx (coalesced)
  hrow[k] = hv;
  __syncthreads();
  if (k < At) {
    float s = 0.0f;
    for (int i = 0; i < Ht; ++i) s += hrow[i] * W_actor[k * Ht + i];
    out[b * At + k] = s;
  } else if (k == At) {
    float s = 0.0f;
    for (int i = 0; i < Ht; ++i) s += hrow[i] * W_critic[i];
    out[Bt * At + b] = s;
  }
}

// ---------------------------------------------------------------------------
extern "C" void kernel_launch(void* const* d_in, const int* in_sizes, int n_in,
                              void* d_out, int out_size, void* d_ws, size_t ws_size,
                              hipStream_t stream) {
  const float* x        = (const float*)d_in[0];
  const float* hx       = (const float*)d_in[1];
  const float* W_ih     = (const float*)d_in[2];
  const float* W_hh     = (const float*)d_in[3];
  const float* W_actor  = (const float*)d_in[4];
  const float* W_critic = (const float*)d_in[5];
  float* out = (float*)d_out;

  // Chunk T so xp scratch (+ h state) fits in workspace: Tc*B*H + B*H floats
  int Tc = 256;
  while (Tc > 1 &&
         ((size_t)Tc * Bt * Ht + (size_t)Bt * Ht) * sizeof(float) > ws_size)
    Tc >>= 1;

  float* xp     = (float*)d_ws;
  float* hstate = xp + (size_t)Tc * Bt * Ht;

  // Seed h state from hx (graph-capture-safe D2D)
  hipMemcpyAsync(hstate, hx, (size_t)Bt * Ht * sizeof(float),
                 hipMemcpyDeviceToDevice, stream);

  const size_t lds1 = (size_t)(256 * WP1 + 64 * XP1) * sizeof(float); // ~169KB
  const size_t lds2 = (size_t)(256 * WP2 + 16 * HP2) * sizeof(float); // ~277KB

  for (int t0 = 0; t0 < Tt; t0 += Tc) {
    xproj_kernel<<<Tc * Bt / 64, 512, lds1, stream>>>(x, W_ih, xp, t0);
    scan_kernel<<<Bt / 16, 256, lds2, stream>>>(xp, W_hh, hstate, Tc);
  }
  heads_kernel<<<Bt, 256, 0, stream>>>(hstate, W_actor, W_critic, out);
}